// CAGKE_fix_54348516164263
// MI455X (gfx1250) — compile-verified
//
#include <hip/hip_runtime.h>
#include <hip/hip_bf16.h>

// ---------------------------------------------------------------------------
// CAGKE on MI455X (gfx1250).
//   psedu[t] = sum_d softmax(w)_d * sum_i spike[i] * (C/sig_d) exp(-(t-1-i)^2/(2 sig_d^2)) + noise[t]
//   out = minmax-normalize(psedu)
// Gaussian support truncated to x in [-128,127]: exact in fp32 (exp underflow).
// Conv done as f32 WMMA (v_wmma_f32_16x16x4_f32) over a 128x256 weighted
// kernel bank (A) and a Toeplitz spike slice in LDS (B).
// ---------------------------------------------------------------------------

typedef __attribute__((ext_vector_type(2))) float v2f;
typedef __attribute__((ext_vector_type(8))) float v8f;

#define DCH   128          // sigma channels (fixed by reference)
#define KWIN  256          // truncated displacement window, x = k-128
#define GAUSS_C 0.39894228f

// Phase A: one block, 256 threads.
// softmax(weight[0:128]) -> fold into Ahat[d][k] = w_d*(C/sig)*exp(-x^2/(2 sig^2))
__global__ __launch_bounds__(256)
void cagke_build_bank(const float* __restrict__ weight,
                      const float* __restrict__ sigma,
                      float* __restrict__ Ahat) {
    __shared__ float wLDS[DCH];
    __shared__ float sgLDS[DCH];
    __shared__ float red[DCH];
    const int tid = threadIdx.x;

    float wv = -3.402823466e38f;
    if (tid < DCH) {
        wv = weight[tid];
        sgLDS[tid] = sigma[tid];
        red[tid] = wv;
    }
    __syncthreads();
    // max-reduce over 128
    for (int s = 64; s > 0; s >>= 1) {
        if (tid < s) red[tid] = fmaxf(red[tid], red[tid + s]);
        __syncthreads();
    }
    const float mx = red[0];
    __syncthreads();
    float e = 0.0f;
    if (tid < DCH) { e = expf(wv - mx); red[tid] = e; }
    __syncthreads();
    for (int s = 64; s > 0; s >>= 1) {
        if (tid < s) red[tid] += red[tid + s];
        __syncthreads();
    }
    const float inv_sum = 1.0f / red[0];
    if (tid < DCH) wLDS[tid] = e * inv_sum;
    __syncthreads();

    // 128*256 = 32768 entries, 128 per thread
    for (int idx = tid; idx < DCH * KWIN; idx += 256) {
        const int d = idx >> 8;
        const int k = idx & (KWIN - 1);
        const float sg = sgLDS[d];
        const float x  = (float)(k - 128);
        const float g  = (GAUSS_C / sg) * expf(-(x * x) / (2.0f * sg * sg));
        Ahat[idx] = wLDS[d] * g;
    }
}

// Phase B: grid = T/128 blocks, 256 threads = 8 waves, 16 outputs per wave.
// One v8f accumulator chains 512 v_wmma_f32_16x16x4_f32 per wave
// (8 d-tiles x 64 K-chunks); final row-sum + lane^16 shuffle gives psedu.
__global__ __launch_bounds__(256)
void cagke_conv_wmma(const float* __restrict__ X,
                     const float* __restrict__ noise,
                     const float* __restrict__ Ahat,
                     float* __restrict__ psedu,
                     int T) {
    __shared__ float sLDS[384];           // spike slice [tB-128, tB+255]
    const int tid = threadIdx.x;
    const int tB  = blockIdx.x * 128;

    // warm the kernel bank (L2/WGP$ resident, 128 KB total)
    __builtin_prefetch(Ahat + tid * 128, 0, 0);

    for (int u = tid; u < 384; u += 256) {
        const int j = tB - 128 + u;
        float v = 0.0f;
        if (j >= 0 && j < T) v = (X[j] > 0.5f) ? 1.0f : 0.0f;  // spike train
        sLDS[u] = v;
    }
    __syncthreads();

    const int wave = tid >> 5;            // 0..7 (wave32)
    const int lane = tid & 31;
    const int n    = lane & 15;           // column (output t within tile) == A row m
    const int half = lane >> 4;           // K-pair select per ISA A/B layouts
    const int base = 16 * wave + n + 255; // sLDS index base; B[k][n] = sLDS[base-k]

    v8f c = {0.f, 0.f, 0.f, 0.f, 0.f, 0.f, 0.f, 0.f};

    for (int kc = 0; kc < 64; ++kc) {
        const int k0 = (kc << 2) + 2 * half;   // lanes 0-15: K0,K1; lanes 16-31: K2,K3
        v2f b;
        b[0] = sLDS[base - k0];
        b[1] = sLDS[base - k0 - 1];
#pragma unroll
        for (int dt = 0; dt < 8; ++dt) {
            // A[m][k0], A[m][k0+1]; row m = n, contiguous even-aligned pair
            const v2f a = *(const v2f*)(Ahat + (dt * 16 + n) * KWIN + k0);
            c = __builtin_amdgcn_wmma_f32_16x16x4_f32(
                    /*neg_a=*/false, a, /*neg_b=*/false, b,
                    /*c_mod=*/(short)0, c, /*reuse_a=*/false, /*reuse_b=*/false);
        }
    }

    // C layout: VGPR v -> rows v (lanes 0-15) / v+8 (lanes 16-31), col = lane&15.
    // Sum all 16 rows of column n (== sum over the 16 d's in each tile).
    float s = c[0] + c[1] + c[2] + c[3] + c[4] + c[5] + c[6] + c[7];
    const float tot = s + __shfl_xor(s, 16, 32);
    if (half == 0) {
        const int t = tB + 16 * wave + n;
        psedu[t] = tot + noise[t];
    }
}

// Phase C: one block; global min/max then normalize.
__global__ __launch_bounds__(256)
void cagke_normalize(const float* __restrict__ psedu,
                     float* __restrict__ out, int T) {
    __shared__ float rmn[256];
    __shared__ float rmx[256];
    const int tid = threadIdx.x;
    float mn = 3.402823466e38f, mx = -3.402823466e38f;
    for (int t = tid; t < T; t += 256) {
        const float v = psedu[t];
        mn = fminf(mn, v);
        mx = fmaxf(mx, v);
    }
    rmn[tid] = mn; rmx[tid] = mx;
    __syncthreads();
    for (int s = 128; s > 0; s >>= 1) {
        if (tid < s) {
            rmn[tid] = fminf(rmn[tid], rmn[tid + s]);
            rmx[tid] = fmaxf(rmx[tid], rmx[tid + s]);
        }
        __syncthreads();
    }
    mn = rmn[0]; mx = rmx[0];
    const float inv = 1.0f / (mx - mn);
    for (int t = tid; t < T; t += 256) {
        out[t] = (psedu[t] - mn) * inv;
    }
}

extern "C" void kernel_launch(void* const* d_in, const int* in_sizes, int n_in,
                              void* d_out, int out_size, void* d_ws, size_t ws_size,
                              hipStream_t stream) {
    // setup_inputs order: X [1,T], weight [1,D], noise [1,T], sigma [D]
    const float* X      = (const float*)d_in[0];
    const float* weight = (const float*)d_in[1];
    const float* noise  = (const float*)d_in[2];
    const float* sigma  = (const float*)d_in[3];
    float* out = (float*)d_out;
    const int T = in_sizes[0];            // 8192

    float* Ahat  = (float*)d_ws;          // 128*256 f32 = 128 KB
    float* psedu = Ahat + DCH * KWIN;     // T f32 = 32 KB

    cagke_build_bank<<<1, 256, 0, stream>>>(weight, sigma, Ahat);
    cagke_conv_wmma<<<T / 128, 256, 0, stream>>>(X, noise, Ahat, psedu, T);
    cagke_normalize<<<1, 256, 0, stream>>>(psedu, out, T);
}